// GraphDilatedConv_82025285419079
// MI455X (gfx1250) — compile-verified
//
#include <hip/hip_runtime.h>
#include <hip/hip_bf16.h>
#include <math.h>

// Problem constants (from reference)
#define BB    4
#define CIN   64
#define TT    4096
#define COUT  64
#define RED   16
#define MAXD  8          // max supported dilation classes (reference uses 2)

typedef __attribute__((ext_vector_type(2))) float v2f;
typedef __attribute__((ext_vector_type(8))) float v8f;

// ---------------------------------------------------------------------------
// K0: repack conv_w [COUT][CIN][3] -> WcT [co][k] with k = kk*64 + ci
// ---------------------------------------------------------------------------
__global__ void repack_convw_kernel(const float* __restrict__ conv_w,
                                    float* __restrict__ wcT) {
    int j = blockIdx.x * 256 + threadIdx.x;
    if (j >= COUT * 192) return;
    int co = j / 192, k = j % 192;
    int ci = k & 63, kk = k >> 6;
    wcT[j] = conv_w[co * 192 + ci * 3 + kk];
}

// ---------------------------------------------------------------------------
// K1: a_j[b,t] = sum_c x[b,c,t] * attn_w[CIN + c]   (a_i term cancels in softmax)
// ---------------------------------------------------------------------------
__global__ void aj_kernel(const float* __restrict__ x,
                          const float* __restrict__ attn_w,
                          float* __restrict__ aj) {
    int blk = blockIdx.x;                       // BB*TT/256 blocks
    int b = blk / (TT / 256);
    int t = (blk % (TT / 256)) * 256 + threadIdx.x;
    const float* xb = x + (size_t)b * CIN * TT + t;
    float s = 0.0f;
#pragma unroll
    for (int c = 0; c < CIN; ++c) s += xb[(size_t)c * TT] * attn_w[CIN + c];
    aj[b * TT + t] = s;
}

// ---------------------------------------------------------------------------
// K2: per (b, r) class statistics: m_r, S_r, V_r[c]
// ---------------------------------------------------------------------------
__global__ void __launch_bounds__(256)
class_stats_kernel(const float* __restrict__ x, const float* __restrict__ aj,
                   const int* __restrict__ dilp,
                   float* __restrict__ wsM, float* __restrict__ wsS,
                   float* __restrict__ wsV) {
    __shared__ float sE[TT];        // exp values for this class (dil=1 worst case)
    __shared__ float sRed[256];
    int dil = *dilp; if (dil < 1) dil = 1; if (dil > MAXD) dil = MAXD;
    int r = blockIdx.x % MAXD;
    int b = blockIdx.x / MAXD;
    if (r >= dil) return;
    int nT = (TT - r + dil - 1) / dil;
    int tid = threadIdx.x;
    const float* ajb = aj + b * TT;

    // phase 1: class max
    float m = -INFINITY;
    for (int i = tid; i < nT; i += 256) m = fmaxf(m, ajb[r + i * dil]);
    sRed[tid] = m; __syncthreads();
    for (int s = 128; s > 0; s >>= 1) {
        if (tid < s) sRed[tid] = fmaxf(sRed[tid], sRed[tid + s]);
        __syncthreads();
    }
    m = sRed[0]; __syncthreads();

    // phase 2: e_t and S_r
    float ssum = 0.0f;
    for (int i = tid; i < nT; i += 256) {
        float e = expf(ajb[r + i * dil] - m);
        sE[i] = e; ssum += e;
    }
    __syncthreads();
    sRed[tid] = ssum; __syncthreads();
    for (int s = 128; s > 0; s >>= 1) {
        if (tid < s) sRed[tid] += sRed[tid + s];
        __syncthreads();
    }
    if (tid == 0) { wsM[b * MAXD + r] = m; wsS[b * MAXD + r] = sRed[0]; }
    __syncthreads();

    // phase 3: V_r[c] = sum_t e_t * x[b,c,t] ; 4 lanes per channel
    int c = tid >> 2, sub = tid & 3;
    const float* xc = x + ((size_t)(b * CIN + c)) * TT;
    float v = 0.0f;
    for (int i = sub; i < nT; i += 4) v += sE[i] * xc[r + i * dil];
    v += __shfl_xor(v, 1);
    v += __shfl_xor(v, 2);
    if (sub == 0) wsV[(b * MAXD + r) * CIN + c] = v;
}

// ---------------------------------------------------------------------------
// K3: fused agg + (agg @ lin_w^T + lin_b) + GLU, via V_WMMA_F32_16X16X4_F32
// One wave = 16 rows (t) x 128 cols; agg computed in-register per A-fragment.
// ---------------------------------------------------------------------------
__global__ void __launch_bounds__(256)
gemm1_glu_kernel(const float* __restrict__ x, const float* __restrict__ aj,
                 const float* __restrict__ lin_w, const float* __restrict__ lin_b,
                 const int* __restrict__ dilp,
                 const float* __restrict__ wsM, const float* __restrict__ wsS,
                 const float* __restrict__ wsV, float* __restrict__ glu) {
    int tid  = threadIdx.x;
    int lane = tid & 31;
    int wave = (blockIdx.x << 3) | (tid >> 5);   // 0..1023
    int b  = wave >> 8;                          // TT/16 = 256 row-tiles per batch
    int t0 = (wave & 255) << 4;
    int ln = lane & 15, hi = lane >> 4;
    int t  = t0 + ln;                            // this lane's matrix row

    int dil = *dilp; if (dil < 1) dil = 1; if (dil > MAXD) dil = MAXD;
    int r = t % dil;
    float m  = wsM[b * MAXD + r];
    float S  = wsS[b * MAXD + r];
    float e  = expf(aj[b * TT + t] - m);
    float inv = 1.0f / (S - e);
    const float* V  = wsV + (b * MAXD + r) * CIN;
    const float* xb = x + (size_t)b * CIN * TT;

    v8f zero = {0.f, 0.f, 0.f, 0.f, 0.f, 0.f, 0.f, 0.f};
    v8f acc[8];
#pragma unroll
    for (int i = 0; i < 8; ++i) acc[i] = zero;

    for (int k0 = 0; k0 < 16; ++k0) {            // K = 64, 4 per WMMA issue
        int kd = (k0 << 2) + (hi << 1);          // this lane's K pair
        float x0 = xb[(size_t)kd * TT + t];
        float x1 = xb[(size_t)(kd + 1) * TT + t];
        v2f a;
        a.x = (V[kd]     - e * x0) * inv;        // agg on the fly (exact softmax)
        a.y = (V[kd + 1] - e * x1) * inv;
#pragma unroll
        for (int ct = 0; ct < 8; ++ct) {
            int n = (ct << 4) + ln;
            v2f bf;
            bf.x = lin_w[n * CIN + kd];
            bf.y = lin_w[n * CIN + kd + 1];
            acc[ct] = __builtin_amdgcn_wmma_f32_16x16x4_f32(
                false, a, false, bf, (short)0, acc[ct], false, false);
        }
    }

    // GLU: col-tile ct (a) pairs with ct+4 (g) on identical lane/VGPR slots
#pragma unroll
    for (int ct = 0; ct < 4; ++ct) {
        int col = (ct << 4) + ln;
        float ba = lin_b[col];
        float bg = lin_b[COUT + col];
#pragma unroll
        for (int rr = 0; rr < 8; ++rr) {
            float av = acc[ct][rr] + ba;
            float gv = acc[ct + 4][rr] + bg;
            float sg = 1.0f / (1.0f + expf(-gv));
            int tr = t0 + rr + (hi << 3);
            glu[((size_t)(b * TT + tr) << 6) + col] = av * sg;
        }
    }
}

// ---------------------------------------------------------------------------
// K4: k=3 pad=1 conv as GEMM [16 t x 192] @ [192 x 64] via WMMA f32, + bias
// glu rows t-1,t,t+1 are contiguous -> A fragment is a plain float2 load.
// ---------------------------------------------------------------------------
__global__ void __launch_bounds__(256)
conv_gemm_kernel(const float* __restrict__ glu, const float* __restrict__ wcT,
                 const float* __restrict__ conv_b, float* __restrict__ convout) {
    int tid  = threadIdx.x;
    int lane = tid & 31;
    int wave = (blockIdx.x << 3) | (tid >> 5);
    int b  = wave >> 8;
    int t0 = (wave & 255) << 4;
    int ln = lane & 15, hi = lane >> 4;
    int t  = t0 + ln;

    v8f zero = {0.f, 0.f, 0.f, 0.f, 0.f, 0.f, 0.f, 0.f};
    v8f acc[4];
#pragma unroll
    for (int i = 0; i < 4; ++i) acc[i] = zero;

    for (int k0 = 0; k0 < 48; ++k0) {            // K = 192
        int kd = (k0 << 2) + (hi << 1);
        v2f a;
        bool zlo = (t == 0)      && (kd < CIN);        // pad row t-1
        bool zhi = (t == TT - 1) && (kd >= 2 * CIN);   // pad row t+1
        if (zlo | zhi) { a.x = 0.0f; a.y = 0.0f; }
        else {
            const float* p = glu + (((size_t)(b * TT + t - 1)) << 6) + kd;
            a.x = p[0]; a.y = p[1];
        }
#pragma unroll
        for (int ct = 0; ct < 4; ++ct) {
            int co = (ct << 4) + ln;
            v2f bf;
            bf.x = wcT[co * 192 + kd];
            bf.y = wcT[co * 192 + kd + 1];
            acc[ct] = __builtin_amdgcn_wmma_f32_16x16x4_f32(
                false, a, false, bf, (short)0, acc[ct], false, false);
        }
    }

#pragma unroll
    for (int ct = 0; ct < 4; ++ct) {
        int co = (ct << 4) + ln;
        float bias = conv_b[co];
#pragma unroll
        for (int rr = 0; rr < 8; ++rr) {
            int tr = t0 + rr + (hi << 3);
            convout[((size_t)(b * TT + tr) << 6) + co] = acc[ct][rr] + bias;
        }
    }
}

// ---------------------------------------------------------------------------
// K5: maxpool(k=2,s=1,left-pad) + transpose [t][c] -> [b][c][t] into d_out
// ---------------------------------------------------------------------------
__global__ void __launch_bounds__(256)
pool_kernel(const float* __restrict__ convout, float* __restrict__ out) {
    __shared__ float lds[65 * 65];               // 65 rows x 64 cols, pitch 65
    int blk = blockIdx.x;                        // BB*TT/64 blocks
    int b  = blk / (TT / 64);
    int t0 = (blk % (TT / 64)) * 64;
    int tid = threadIdx.x;
    for (int idx = tid; idx < 65 * 64; idx += 256) {
        int row = idx >> 6, c = idx & 63;
        int tt = t0 - 1 + row; if (tt < 0) tt = 0;   // pool[0] = out[0]
        lds[row * 65 + c] = convout[((size_t)(b * TT + tt) << 6) + c];
    }
    __syncthreads();
#pragma unroll
    for (int it = 0; it < 16; ++it) {
        int j = it * 256 + tid;
        int c = j >> 6, tt = j & 63;
        float v = fmaxf(lds[(tt + 1) * 65 + c], lds[tt * 65 + c]);
        out[((size_t)(b * COUT + c)) * TT + t0 + tt] = v;
    }
}

// ---------------------------------------------------------------------------
// K6: mean over T per (b,c)
// ---------------------------------------------------------------------------
__global__ void mean_kernel(const float* __restrict__ out, float* __restrict__ mean) {
    __shared__ float red[256];
    int bc = blockIdx.x;
    const float* p = out + (size_t)bc * TT;
    float s = 0.0f;
    for (int i = threadIdx.x; i < TT; i += 256) s += p[i];
    red[threadIdx.x] = s; __syncthreads();
    for (int st = 128; st > 0; st >>= 1) {
        if (threadIdx.x < st) red[threadIdx.x] += red[threadIdx.x + st];
        __syncthreads();
    }
    if (threadIdx.x == 0) mean[bc] = red[0] * (1.0f / TT);
}

// ---------------------------------------------------------------------------
// K7: squeeze-excite MLP: relu(mean@W1^T+b1) -> sigmoid(@W2^T+b2)
// ---------------------------------------------------------------------------
__global__ void se_kernel(const float* __restrict__ mean,
                          const float* __restrict__ w1, const float* __restrict__ b1,
                          const float* __restrict__ w2, const float* __restrict__ b2,
                          float* __restrict__ se) {
    __shared__ float h[BB * (COUT / RED)];
    int tid = threadIdx.x;
    if (tid < BB * (COUT / RED)) {
        int b = tid >> 2, j = tid & 3;
        float s = b1[j];
        for (int c = 0; c < COUT; ++c) s += mean[b * COUT + c] * w1[j * COUT + c];
        h[tid] = fmaxf(s, 0.0f);
    }
    __syncthreads();
    int b = tid >> 6, c = tid & 63;
    float s = b2[c];
#pragma unroll
    for (int j = 0; j < COUT / RED; ++j)
        s += h[b * (COUT / RED) + j] * w2[c * (COUT / RED) + j];
    se[tid] = 1.0f / (1.0f + expf(-s));
}

// ---------------------------------------------------------------------------
// K8: scale d_out in place by se[b*COUT+c]  (index/T == b*COUT+c)
// ---------------------------------------------------------------------------
__global__ void scale_kernel(float* __restrict__ out, const float* __restrict__ se) {
    size_t i = (size_t)blockIdx.x * 256 + threadIdx.x;
    out[i] *= se[i / TT];
}

// ---------------------------------------------------------------------------
extern "C" void kernel_launch(void* const* d_in, const int* in_sizes, int n_in,
                              void* d_out, int out_size, void* d_ws, size_t ws_size,
                              hipStream_t stream) {
    const float* x      = (const float*)d_in[0];
    const float* attn_w = (const float*)d_in[1];
    // attn_b (d_in[2]) cancels in the softmax and is unused.
    const float* lin_w  = (const float*)d_in[3];
    const float* lin_b  = (const float*)d_in[4];
    const float* conv_w = (const float*)d_in[5];
    const float* conv_b = (const float*)d_in[6];
    const float* se_w1  = (const float*)d_in[7];
    const float* se_b1  = (const float*)d_in[8];
    const float* se_w2  = (const float*)d_in[9];
    const float* se_b2  = (const float*)d_in[10];
    const int*   dil    = (const int*)d_in[11];
    float* out = (float*)d_out;

    float* ws = (float*)d_ws;
    float* wsAJ   = ws;                     // BB*TT            = 16384
    float* wsM    = wsAJ + BB * TT;         // BB*MAXD          = 32
    float* wsS    = wsM + BB * MAXD;        // 32
    float* wsV    = wsS + BB * MAXD;        // BB*MAXD*CIN      = 2048
    float* wsWcT  = wsV + BB * MAXD * CIN;  // COUT*192         = 12288
    float* wsMean = wsWcT + COUT * 192;     // BB*COUT          = 256
    float* wsSE   = wsMean + BB * COUT;     // 256
    float* wsGLU  = ws + 32768;             // BB*TT*COUT       = 1048576
    float* wsConv = wsGLU + (size_t)BB * TT * COUT;  // 1048576

    repack_convw_kernel<<<(COUT * 192 + 255) / 256, 256, 0, stream>>>(conv_w, wsWcT);
    aj_kernel<<<BB * TT / 256, 256, 0, stream>>>(x, attn_w, wsAJ);
    class_stats_kernel<<<BB * MAXD, 256, 0, stream>>>(x, wsAJ, dil, wsM, wsS, wsV);
    gemm1_glu_kernel<<<BB * TT / 16 / 8, 256, 0, stream>>>(
        x, wsAJ, lin_w, lin_b, dil, wsM, wsS, wsV, wsGLU);
    conv_gemm_kernel<<<BB * TT / 16 / 8, 256, 0, stream>>>(wsGLU, wsWcT, conv_b, wsConv);
    pool_kernel<<<BB * TT / 64, 256, 0, stream>>>(wsConv, out);
    mean_kernel<<<BB * COUT, 256, 0, stream>>>(out, wsMean);
    se_kernel<<<1, 256, 0, stream>>>(wsMean, se_w1, se_b1, se_w2, se_b2, wsSE);
    scale_kernel<<<(size_t)BB * COUT * TT / 256, 256, 0, stream>>>(out, wsSE);
}